// QuantizedLatent_87900800680049
// MI455X (gfx1250) — compile-verified
//
#include <hip/hip_runtime.h>
#include <hip/hip_bf16.h>
#include <stdint.h>

// FSQ-style per-latent scalar quantization, L=8.39M latents, V=16 codes/latent.
// Memory-bound: 84 B/latent -> ~705 MB one-pass stream -> ~30us at 23.3 TB/s.
// CDNA5 path: async global->LDS staging of codebook rows (coalesced 512B/wave
// transfers on ASYNCcnt) + non-temporal cache hints (stream >> 192MB L2).

#define TILE 256   // latents per block (8 wave32s)
#define ROWB 64    // bytes per codebook row (16 * f32)

typedef float vfloat4 __attribute__((ext_vector_type(4)));

__global__ __launch_bounds__(TILE) void QuantizedLatent_fsq_kernel(
    const float* __restrict__ x,
    const float* __restrict__ values,   // [n, 16] row-major f32
    float* __restrict__ out,            // [x | quantized | z_hat | indices], 4*n f32
    int n)
{
    // 256 rows * 64B = 16KB staged codebook tile
    __shared__ vfloat4 tile[TILE * 4];

    const int tid  = threadIdx.x;
    const int t0   = blockIdx.x * TILE;
    const int i    = t0 + tid;
    const int wave = tid >> 5;
    const int lane = tid & 31;

    const bool fullTile = (t0 + TILE) <= n;

    float v[16];

    if (fullTile) {
        // Wave w owns tile bytes [w*2048, (w+1)*2048): exactly the 32 rows
        // (w*32+lane) that wave w's lanes consume below -> no barrier needed,
        // only this wave's own ASYNCcnt.
        const char*    gbase = (const char*)values + ((size_t)t0 * ROWB);
        const uint32_t lbase = (uint32_t)(uintptr_t)tile;  // LDS byte offset
        #pragma unroll
        for (int k = 0; k < 4; ++k) {
            const uint32_t off   = (uint32_t)(wave * 2048 + k * 512 + lane * 16);
            const uint32_t laddr = lbase + off;
            const uint64_t gaddr = (uint64_t)(uintptr_t)(gbase + off);
            // Fully coalesced 512B per wave per instruction; NT: no L2 reuse.
            asm volatile("global_load_async_to_lds_b128 %0, %1, off th:TH_LOAD_NT"
                         :: "v"(laddr), "v"(gaddr)
                         : "memory");
        }
        asm volatile("s_wait_asynccnt 0x0" ::: "memory");

        const vfloat4 q0 = tile[tid * 4 + 0];
        const vfloat4 q1 = tile[tid * 4 + 1];
        const vfloat4 q2 = tile[tid * 4 + 2];
        const vfloat4 q3 = tile[tid * 4 + 3];
        v[ 0]=q0.x; v[ 1]=q0.y; v[ 2]=q0.z; v[ 3]=q0.w;
        v[ 4]=q1.x; v[ 5]=q1.y; v[ 6]=q1.z; v[ 7]=q1.w;
        v[ 8]=q2.x; v[ 9]=q2.y; v[10]=q2.z; v[11]=q2.w;
        v[12]=q3.x; v[13]=q3.y; v[14]=q3.z; v[15]=q3.w;
    } else if (i < n) {
        // Tail block (not hit for n = 8388608): plain NT b128 loads.
        const vfloat4* row = (const vfloat4*)(values + (size_t)i * 16);
        const vfloat4 q0 = __builtin_nontemporal_load(row + 0);
        const vfloat4 q1 = __builtin_nontemporal_load(row + 1);
        const vfloat4 q2 = __builtin_nontemporal_load(row + 2);
        const vfloat4 q3 = __builtin_nontemporal_load(row + 3);
        v[ 0]=q0.x; v[ 1]=q0.y; v[ 2]=q0.z; v[ 3]=q0.w;
        v[ 4]=q1.x; v[ 5]=q1.y; v[ 6]=q1.z; v[ 7]=q1.w;
        v[ 8]=q2.x; v[ 9]=q2.y; v[10]=q2.z; v[11]=q2.w;
        v[12]=q3.x; v[13]=q3.y; v[14]=q3.z; v[15]=q3.w;
    }

    if (i < n) {
        const float xi = __builtin_nontemporal_load(x + i);

        // First-occurrence argmin of |xi - v[j]| (strict '<' like jnp.argmin).
        float bd = fabsf(xi - v[0]);
        float bv = v[0];
        int   bi = 0;
        #pragma unroll
        for (int j = 1; j < 16; ++j) {
            const float d = fabsf(xi - v[j]);
            if (d < bd) { bd = d; bv = v[j]; bi = j; }
        }

        const size_t N = (size_t)n;
        // z_hat = x + (q - x) evaluated in fp32 (NOT folded to q) to match
        // the reference's straight-through-estimator arithmetic bit-exactly.
        const float zhat = xi + (bv - xi);

        __builtin_nontemporal_store(xi,         out + i);          // z_continuous
        __builtin_nontemporal_store(bv,         out + N + i);      // z_quantized
        __builtin_nontemporal_store(zhat,       out + 2 * N + i);  // z_hat
        __builtin_nontemporal_store((float)bi,  out + 3 * N + i);  // z_indices
    }
}

extern "C" void kernel_launch(void* const* d_in, const int* in_sizes, int n_in,
                              void* d_out, int out_size, void* d_ws, size_t ws_size,
                              hipStream_t stream) {
    const float* x      = (const float*)d_in[0];
    const float* values = (const float*)d_in[1];
    float*       out    = (float*)d_out;
    const int n = in_sizes[0];

    const int blocks = (n + TILE - 1) / TILE;
    QuantizedLatent_fsq_kernel<<<blocks, TILE, 0, stream>>>(x, values, out, n);
}